// SignatureTransform_48473000902865
// MI455X (gfx1250) — compile-verified
//
#include <hip/hip_runtime.h>
#include <hip/hip_bf16.h>

// Path-signature (depth 3, time-augmented, normalized) for (8192, 2048, 4) f32.
// Per-batch GEMM  D[16x32] += A[16x4] x B[4x32]  over K=time using
// V_WMMA_F32_16X16X4_F32 (two 16-wide N tiles).  A = inc channels (rows 0..4
// valid), B columns = [left_j*left_k (25) | left_j (5, sig2 via ones channel) |
// junk].  Staging uses GLOBAL_LOAD_ASYNC_TO_LDS_B128 (ASYNCcnt).

typedef __attribute__((ext_vector_type(2))) float v2f;
typedef __attribute__((ext_vector_type(8))) float v8f;

#define BATCH   8192
#define LPATH   2048
#define NCH0    4
#define CH      5              // channels after time augment
#define TINC    (LPATH - 1)    // 2047 increments
#define TPAD    LPATH          // pad to 2048; slot 2047 zero-filled
#define K4      (TPAD / 4)     // 512 k-steps of 4 timesteps
#define NWAVES  8
#define BLOCK   (NWAVES * 32)
#define OUTC    155
#define LSTRIDE 6              // left array: 5 channels + constant 1.0
#define ISTRIDE 5

#define SMEM_FLOATS (LPATH*NCH0 + TPAD*LSTRIDE + TPAD*ISTRIDE + NWAVES*2*256 + (OUTC+1))

__global__ __launch_bounds__(BLOCK)
void sig_wmma_kernel(const float* __restrict__ path, float* __restrict__ out) {
    extern __shared__ float smem[];
    float* raw   = smem;                       // 2048*4
    float* leftS = raw   + LPATH * NCH0;       // 2048*6
    float* incS  = leftS + TPAD * LSTRIDE;     // 2048*5
    float* part  = incS  + TPAD * ISTRIDE;     // 8 waves * 2 tiles * 256
    float* sigb  = part  + NWAVES * 2 * 256;   // 155 + 1 (inv-norm)

    const int b   = blockIdx.x;
    const int tid = threadIdx.x;
    const float* pb = path + (size_t)b * (LPATH * NCH0);

    // ---- stage raw path[b] into LDS via async DMA (no VGPR round-trip) ----
    // LDS flat addressing truncates to addr[31:0], so the generic pointer's
    // low 32 bits are the LDS byte offset for the async-load VDST operand.
    {
        const unsigned rawBase = (unsigned)(unsigned long long)(uintptr_t)raw;
        for (int i = tid; i < LPATH * NCH0 / 4; i += BLOCK) {
            unsigned      ldsOff = rawBase + (unsigned)i * 16u;
            const float4* gsrc   = ((const float4*)pb) + i;
            asm volatile("global_load_async_to_lds_b128 %0, %1, off"
                         :: "v"(ldsOff), "v"(gsrc) : "memory");
        }
        asm volatile("s_wait_asynccnt 0" ::: "memory");
    }
    __syncthreads();

    // ---- build left/inc arrays with time channel; zero-pad t == 2047 ----
    const float invT = 1.0f / (float)TINC;     // increment of the time channel
    for (int t = tid; t < TPAD; t += BLOCK) {
        const bool valid = (t < TINC);
        leftS[t*LSTRIDE + 0] = valid ? (float)t * invT : 0.0f;
        incS [t*ISTRIDE + 0] = valid ? invT            : 0.0f;
#pragma unroll
        for (int c = 0; c < NCH0; ++c) {
            float cur = raw[t * NCH0 + c];
            float nxt = valid ? raw[(t + 1) * NCH0 + c] : cur;
            leftS[t*LSTRIDE + 1 + c] = valid ? cur : 0.0f;
            incS [t*ISTRIDE + 1 + c] = nxt - cur;
        }
        leftS[t*LSTRIDE + 5] = valid ? 1.0f : 0.0f;   // ones channel
    }
    __syncthreads();

    // ---- per-lane WMMA operand metadata ----
    const int lane = tid & 31;
    const int wave = tid >> 5;
    const int lnm  = lane & 15;
    const int half = lane >> 4;

    const int iA = lnm % CH;                 // A row channel (rows 5..15 -> junk D rows)
    const int j0 = lnm / 5, k0 = lnm % 5;    // tile0 columns: jk = 0..15
    int o1, o2;                              // tile1 channel offsets into leftS
    {
        const int n1 = 16 + lnm;
        if (n1 < 25)      { o1 = n1 / 5;  o2 = n1 % 5; } // jk = 16..24
        else if (n1 < 30) { o1 = n1 - 25; o2 = 5;      } // sig2: left_c * 1
        else              { o1 = 5;       o2 = 5;      } // junk cols 30,31: 1*1
    }

    // Two independent accumulator chains per tile (k-steps kk and kk+NWAVES) so the
    // scheduler can overlap one chain's DS loads with the other's WMMAs.
    v8f acc0a = {0.f,0.f,0.f,0.f,0.f,0.f,0.f,0.f};
    v8f acc1a = acc0a, acc0b = acc0a, acc1b = acc0a;

    auto step = [&](int kk, v8f& a0, v8f& a1) {
        const int ta = kk * 4 + half * 2;    // this lane supplies K slots {ta, ta+1}
        const float* La = leftS + ta * LSTRIDE;
        const float* Lb = La + LSTRIDE;
        const float* Ia = incS + ta * ISTRIDE;

        v2f A;
        A.x = Ia[iA];
        A.y = Ia[ISTRIDE + iA];

        v2f B0;
        B0.x = La[j0] * La[k0];
        B0.y = Lb[j0] * Lb[k0];
        a0 = __builtin_amdgcn_wmma_f32_16x16x4_f32(false, A, false, B0,
                                                   (short)0, a0, false, false);
        v2f B1;
        B1.x = La[o1] * La[o2];
        B1.y = Lb[o1] * Lb[o2];
        a1 = __builtin_amdgcn_wmma_f32_16x16x4_f32(false, A, false, B1,
                                                   (short)0, a1, false, false);
    };

    // K4 = 512 is divisible by 2*NWAVES = 16, so no tail.
    for (int kk = wave; kk < K4; kk += 2 * NWAVES) {
        step(kk,          acc0a, acc1a);
        step(kk + NWAVES, acc0b, acc1b);
    }
    v8f acc0 = acc0a + acc0b;
    v8f acc1 = acc1a + acc1b;

    // ---- dump accumulators: element (M,N) -> vgpr v=M%8, lane = N + 16*(M/8) ----
#pragma unroll
    for (int v = 0; v < 8; ++v) {
        part[(wave*2 + 0)*256 + v*32 + lane] = acc0[v];
        part[(wave*2 + 1)*256 + v*32 + lane] = acc1[v];
    }
    __syncthreads();

    // ---- combine wave partials into the 155-vector ----
    if (tid < OUTC) {
        float val;
        if (tid < 5) {
            // sig1 = path[-1] - path[0]; time channel: 1 - 0
            val = (tid == 0) ? 1.0f
                 : (raw[(LPATH - 1) * NCH0 + (tid - 1)] - raw[tid - 1]);
        } else if (tid < 30) {
            const int i = (tid - 5) / 5, j = (tid - 5) % 5;
            const int N = 9 + j;             // tile1 local col for 25+j
            float s = 0.f;
            for (int w = 0; w < NWAVES; ++w)
                s += part[(w*2 + 1)*256 + i*32 + N];   // i<8 -> v=i, lane=N
            val = s;
        } else {
            const int r = tid - 30;
            const int i = r / 25, jk = r % 25;
            const int tile = (jk < 16) ? 0 : 1;
            const int N    = (jk < 16) ? jk : (jk - 16);
            float s = 0.f;
            for (int w = 0; w < NWAVES; ++w)
                s += part[(w*2 + tile)*256 + i*32 + N];
            val = s * (1.0f / (float)LPATH);           // sig3 / L
        }
        sigb[tid] = val;
    }
    __syncthreads();
    if (tid == 0) {
        float ssq = 0.f;
        for (int e = 0; e < OUTC; ++e) ssq += sigb[e] * sigb[e];
        sigb[OUTC] = 1.0f / (sqrtf(ssq) + 1e-8f);
    }
    __syncthreads();
    if (tid < OUTC) {
        out[(size_t)b * OUTC + tid] = sigb[tid] * sigb[OUTC];
    }
}

extern "C" void kernel_launch(void* const* d_in, const int* in_sizes, int n_in,
                              void* d_out, int out_size, void* d_ws, size_t ws_size,
                              hipStream_t stream) {
    (void)in_sizes; (void)n_in; (void)d_ws; (void)ws_size; (void)out_size;
    const float* path = (const float*)d_in[0];
    float* out = (float*)d_out;

    const size_t smem = (size_t)SMEM_FLOATS * sizeof(float);   // ~140 KB (<320 KB/WGP)
    hipFuncSetAttribute((const void*)sig_wmma_kernel,
                        hipFuncAttributeMaxDynamicSharedMemorySize, (int)smem);
    sig_wmma_kernel<<<BATCH, BLOCK, smem, stream>>>(path, out);
}